// QGate_65481071399620
// MI455X (gfx1250) — compile-verified
//
#include <hip/hip_runtime.h>
#include <hip/hip_bf16.h>

// ---------------------------------------------------------------------------
// Quantum circuit simulator (14 wires, depth 8, batch 1024) for MI455X gfx1250
//
// Key structure:
//  * RX embedding on |0..0> == product state -> direct closed-form init.
//    Each amplitude is (prod of cos/sin) * (-i)^popcount -> purely re or im.
//  * RY layers + CNOT rings are REAL -> evolve re/im vectors independently.
//  * 4 RY gates on 4 adjacent index bits = one 16x16 real unitary ->
//    applied as 16x16 x (16 x 1024) GEMM via V_WMMA_F32_16X16X4_F32
//    (four chained K=4 WMMAs per 16x16x16 tile), full f32 precision.
//  * CNOT ring = GF(2)-linear index permutation sigma; applied in-place
//    per layer with register staging + scatter (sigma(k) = XOR of basis images).
//  * Whole statevector (re+im = 128 KB) lives in LDS for the entire circuit;
//    only 1024x14 floats in, 1024x14 floats out touch HBM.
// ---------------------------------------------------------------------------

typedef __attribute__((ext_vector_type(2))) float v2f;
typedef __attribute__((ext_vector_type(8))) float v8f;

#define N_WIRES  14
#define DEPTH    8
#define BATCH    1024
#define NSTATE   16384       // 2^14
#define NTHREADS 256         // 8 wave32 per workgroup
#define EPT      64          // NSTATE / NTHREADS elements per thread
#define UPITCH   17          // 16 + 1 pad (17 coprime to 64 banks)

// LDS float layout offsets
#define OFF_RE   0
#define OFF_IM   (OFF_RE + NSTATE)
#define OFF_U    (OFF_IM + NSTATE)                 // 4 * 16 * UPITCH = 1088
#define OFF_CT   (OFF_U + 4 * 16 * UPITCH)         // DEPTH*N_WIRES = 112
#define OFF_ST   (OFF_CT + DEPTH * N_WIRES)        // 112
#define OFF_CH   (OFF_ST + DEPTH * N_WIRES)        // 14
#define OFF_SH   (OFF_CH + N_WIRES)                // 14
#define OFF_Z    (OFF_SH + N_WIRES)                // 14
#define SMEM_FLOATS (OFF_Z + N_WIRES)
#define SMEM_BYTES  (SMEM_FLOATS * 4)              // ~136.5 KB

// state index for a 16x16 unitary acting on index bits [p+3 .. p]:
// k = h*16*W + y*W + w, W = 2^p; column id c = h*W + w.
__device__ __forceinline__ int sidx(int y, int c, int p) {
    int W = 1 << p;
    int h = c >> p;
    int w = c & (W - 1);
    return (h << (p + 4)) + (y << p) + w;
}

__global__ void __launch_bounds__(NTHREADS)
qsim_kernel(const float* __restrict__ inp,    // (BATCH, N_WIRES)
            const float* __restrict__ theta,  // (DEPTH, N_WIRES)
            float* __restrict__ out)          // (BATCH, N_WIRES)
{
    extern __shared__ float smem[];
    float* reS  = smem + OFF_RE;
    float* imS  = smem + OFF_IM;
    float* Umat = smem + OFF_U;
    float* ctL  = smem + OFF_CT;
    float* stL  = smem + OFF_ST;
    float* chL  = smem + OFF_CH;
    float* shL  = smem + OFF_SH;
    float* zsum = smem + OFF_Z;

    const int tid  = threadIdx.x;
    const int lane = tid & 31;     // wave32 (gfx1250)
    const int wid  = tid >> 5;     // 8 waves
    const int b    = blockIdx.x;

    // ---- trig tables (theta shared across batch; inp per-sample) ----------
    if (tid < DEPTH * N_WIRES) {
        float th = 0.5f * theta[tid];
        ctL[tid] = __cosf(th);
        stL[tid] = __sinf(th);
    }
    if (tid < N_WIRES) {
        float x = 0.5f * inp[b * N_WIRES + tid];
        chL[tid] = __cosf(x);
        shL[tid] = __sinf(x);
        zsum[tid] = 0.0f;
    }
    __syncthreads();

    // ---- init: product state from RX embedding ----------------------------
    // amp(k) = prod_i (bit?sin:cos) * (-i)^popcount(k); wire i <-> bit (13-i)
    for (int j = 0; j < EPT; ++j) {
        int k = j * NTHREADS + tid;
        float mag = 1.0f;
#pragma unroll
        for (int bp = 0; bp < N_WIRES; ++bp) {
            int w = N_WIRES - 1 - bp;
            mag *= ((k >> bp) & 1) ? shL[w] : chL[w];
        }
        int pc = __popc(k) & 3;
        float re = 0.0f, im = 0.0f;
        if      (pc == 0) re =  mag;
        else if (pc == 1) im = -mag;
        else if (pc == 2) re = -mag;
        else              im =  mag;
        reS[k] = re;
        imS[k] = im;
    }

    // ---- CNOT-ring permutation: sigma is GF(2)-linear; basis images -------
    int sb[N_WIRES];
#pragma unroll
    for (int bp = 0; bp < N_WIRES; ++bp) {
        int x = 1 << bp;
#pragma unroll
        for (int i = 0; i < N_WIRES - 1; ++i)   // CNOT(i, i+1): bits 13-i -> 12-i
            if ((x >> (13 - i)) & 1) x ^= 1 << (12 - i);
        if (x & 1) x ^= 1 << 13;                // CNOT(13, 0)
        sb[bp] = x;
    }
    int sig_t = 0;                              // sigma contribution of tid bits
#pragma unroll
    for (int bp = 0; bp < 8; ++bp)
        if ((tid >> bp) & 1) sig_t ^= sb[bp];

    const int GP[4] = {10, 6, 2, 0};            // bit position p of each group

    // ---- variational layers -----------------------------------------------
    for (int d = 0; d < DEPTH; ++d) {
        __syncthreads();
        // Build the four 16x16 group unitaries (tensor products of RY).
        // Groups 0..2: wires 4g..4g+3. Group 3: I (x) I (x) RY12 (x) RY13.
        {
            int m = tid >> 4, y = tid & 15;
#pragma unroll
            for (int g = 0; g < 4; ++g) {
                float prod = 1.0f;
#pragma unroll
                for (int jj = 0; jj < 4; ++jj) {
                    int xb = (m >> (3 - jj)) & 1;
                    int yb = (y >> (3 - jj)) & 1;
                    int wire = (g < 3) ? (4 * g + jj) : ((jj >= 2) ? (10 + jj) : -1);
                    float f;
                    if (wire >= 0) {
                        float c = ctL[d * N_WIRES + wire];
                        float s = stL[d * N_WIRES + wire];
                        // RY = [[c, -s], [s, c]]
                        f = xb ? (yb ? c : s) : (yb ? -s : c);
                    } else {
                        f = (xb == yb) ? 1.0f : 0.0f;
                    }
                    prod *= f;
                }
                Umat[(g * 16 + m) * UPITCH + y] = prod;
            }
        }
        __syncthreads();

        // Apply each group unitary to both real vectors via f32 WMMA.
#pragma unroll
        for (int g = 0; g < 4; ++g) {
            const int p  = GP[g];
            const int m  = lane & 15;   // row for A, column-in-tile for B/D
            const int kh = lane >> 4;   // K-half select
            // A fragment (16x4 f32): lanes 0-15 K=0,1; lanes 16-31 K=2,3
            v2f afrag[4];
#pragma unroll
            for (int jc = 0; jc < 4; ++jc) {
                afrag[jc].x = Umat[(g * 16 + m) * UPITCH + (4 * jc + 2 * kh)];
                afrag[jc].y = Umat[(g * 16 + m) * UPITCH + (4 * jc + 2 * kh + 1)];
            }
            // 64 column-tiles x {re, im} = 128 tile-ops over 8 waves.
            // Tiles are disjoint -> no intra-group barrier needed.
            for (int t = wid; t < 128; t += 8) {
                float* S = (t & 1) ? imS : reS;
                const int col = ((t >> 1) << 4) + m;
                v8f acc = {0.f, 0.f, 0.f, 0.f, 0.f, 0.f, 0.f, 0.f};
#pragma unroll
                for (int jc = 0; jc < 4; ++jc) {
                    int k0 = 4 * jc + 2 * kh;
                    v2f bfrag;
                    bfrag.x = S[sidx(k0,     col, p)];
                    bfrag.y = S[sidx(k0 + 1, col, p)];
                    // D(16x16) += A(16x4) x B(4x16), f32 in/out
                    acc = __builtin_amdgcn_wmma_f32_16x16x4_f32(
                        false, afrag[jc], false, bfrag, (short)0, acc,
                        false, false);
                }
                // D layout: VGPR r holds rows r (lanes 0-15) / r+8 (lanes 16-31)
#pragma unroll
                for (int r = 0; r < 8; ++r)
                    S[sidx(r + 8 * kh, col, p)] = acc[r];
            }
            __syncthreads();
        }

        // CNOT ring: in-place permutation, register-staged (two passes to cap
        // staging at 64 VGPRs/lane). k = (j<<8)|tid; sigma(k)=sig_t ^ sigma(j<<8).
#pragma unroll
        for (int pass = 0; pass < 2; ++pass) {
            float* S = pass ? imS : reS;
            float rg[EPT];
#pragma unroll
            for (int j = 0; j < EPT; ++j)
                rg[j] = S[j * NTHREADS + tid];
            __syncthreads();
#pragma unroll
            for (int j = 0; j < EPT; ++j) {
                int sj = sig_t;
#pragma unroll
                for (int bp = 0; bp < 6; ++bp)
                    if ((j >> bp) & 1) sj ^= sb[8 + bp];
                S[sj] = rg[j];
            }
            __syncthreads();
        }
    }

    // ---- readout: <Z_i> = sum_k p(k) * (1 - 2*bit_i(k)) --------------------
    float z[N_WIRES];
#pragma unroll
    for (int i = 0; i < N_WIRES; ++i) z[i] = 0.0f;
    for (int j = 0; j < EPT; ++j) {
        int k = j * NTHREADS + tid;
        float re = reS[k], im = imS[k];
        float pr = re * re + im * im;
#pragma unroll
        for (int i = 0; i < N_WIRES; ++i)
            z[i] += ((k >> (13 - i)) & 1) ? -pr : pr;
    }
#pragma unroll
    for (int i = 0; i < N_WIRES; ++i)
        atomicAdd(&zsum[i], z[i]);   // ds_add_f32
    __syncthreads();
    if (tid < N_WIRES) out[b * N_WIRES + tid] = zsum[tid];
}

extern "C" void kernel_launch(void* const* d_in, const int* in_sizes, int n_in,
                              void* d_out, int out_size, void* d_ws, size_t ws_size,
                              hipStream_t stream) {
    (void)in_sizes; (void)n_in; (void)d_ws; (void)ws_size; (void)out_size;
    const float* inp   = (const float*)d_in[0];   // (1024, 14) f32
    const float* theta = (const float*)d_in[1];   // (8, 14)    f32
    float* out = (float*)d_out;                   // (1024, 14) f32

    // Allow >64KB dynamic LDS (capture-safe: not a stream operation).
    (void)hipFuncSetAttribute((const void*)qsim_kernel,
                              hipFuncAttributeMaxDynamicSharedMemorySize,
                              SMEM_BYTES);

    qsim_kernel<<<BATCH, NTHREADS, SMEM_BYTES, stream>>>(inp, theta, out);
}